// ScigptMoeDecoderLayerPP_21474836480726
// MI455X (gfx1250) — compile-verified
//
#include <hip/hip_runtime.h>
#include <hip/hip_bf16.h>
#include <math.h>
#include <stdint.h>

typedef float v2f __attribute__((ext_vector_type(2)));
typedef float v8f __attribute__((ext_vector_type(8)));

constexpr int H   = 2048;
constexpr int NH  = 32;
constexpr int NKV = 8;
constexpr int HD  = 64;
constexpr int I   = 4096;
constexpr int E   = 8;
constexpr int S   = 2048;
constexpr int NL  = 4;
constexpr int KVW = NKV * HD;   // 512
constexpr float EPS = 1e-5f;

enum { EP_STORE = 0, EP_ADD = 1, EP_SILU_MUL = 2, EP_SCALE_ACC = 3 };

__device__ __forceinline__ v8f wmma4(v2f a, v2f b, v8f c) {
    // V_WMMA_F32_16X16X4_F32 : D = A(16x4) * B(4x16) + C, f32 throughout
    return __builtin_amdgcn_wmma_f32_16x16x4_f32(false, a, false, b, (short)0, c,
                                                 false, false);
}

// LDS byte offset of a __shared__ object: LDS aperture occupies addr[63:32],
// so the low 32 bits of the flat pointer are the LDS address.
__device__ __forceinline__ uint32_t lds_off_of(const void* p) {
    return (uint32_t)(uintptr_t)p;
}

// GLOBAL_LOAD_ASYNC_TO_LDS_B128: per-lane 16B global -> LDS, tracked by ASYNCcnt.
__device__ __forceinline__ void async_ld_b128(uint32_t ldsoff, const float* g) {
    asm volatile("global_load_async_to_lds_b128 %0, %1, off"
                 :: "v"(ldsoff), "v"(g) : "memory");
}

__device__ __forceinline__ void wait_async0() {
    asm volatile("s_wait_asynccnt 0x0" ::: "memory");
}

// ---------------------------------------------------------------------------
// Tiled WMMA GEMM:  C[M,N] = A[M,K] * B[N,K]^T   (A, B row-major, K inner)
// block = 128 threads (4 waves), tile 64(M) x 64(N), BK = 16.
// Double-buffered LDS filled with async global->LDS copies (ASYNCcnt),
// overlapped with the 16 WMMAs of the in-flight tile.
// Epilogues: store | +aux (residual) | silu(aux)*val | C += rowscale[row]*val
// ---------------------------------------------------------------------------
__global__ __launch_bounds__(128)
void wmma_gemm_kernel(const float* __restrict__ A, const float* __restrict__ B,
                      float* __restrict__ C, const float* __restrict__ aux,
                      const float* __restrict__ rowscale, int rs_stride,
                      int M, int N, int K, int mode)
{
    // stride 20 floats: 16B-aligned b128 stores, conflict-free float2 frag reads
    __shared__ float As[2][64][20];
    __shared__ float Bs[2][64][20];

    const int tid  = threadIdx.x;
    const int lane = tid & 31;
    const int wave = tid >> 5;
    const int bm   = blockIdx.y * 64;
    const int bn   = blockIdx.x * 64;
    const int mr   = lane & 15;          // row-in-tile for A frag / col for B frag
    const int kh   = (lane >> 4) << 1;   // K pair offset: lanes 0-15 -> 0, 16-31 -> 2

    v8f acc[4];
#pragma unroll
    for (int j = 0; j < 4; ++j) { v8f z = {0,0,0,0,0,0,0,0}; acc[j] = z; }

    // this thread's two staging slots within a 64x16 tile
    const int row0 = (tid + 0)   >> 2;           // 0..31
    const int c40  = ((tid + 0) & 3) << 2;       // 0,4,8,12
    const int row1 = (tid + 128) >> 2;           // 32..63
    const int c41  = ((tid + 128) & 3) << 2;

    // prologue: prefetch tile 0 into buffer 0
    async_ld_b128(lds_off_of(&As[0][row0][c40]), A + (size_t)(bm + row0) * K + c40);
    async_ld_b128(lds_off_of(&Bs[0][row0][c40]), B + (size_t)(bn + row0) * K + c40);
    async_ld_b128(lds_off_of(&As[0][row1][c41]), A + (size_t)(bm + row1) * K + c41);
    async_ld_b128(lds_off_of(&Bs[0][row1][c41]), B + (size_t)(bn + row1) * K + c41);

    int buf = 0;
    for (int kt = 0; kt < K; kt += 16) {
        wait_async0();          // this wave's copies for buffer `buf` are in LDS
        __syncthreads();        // every wave's copies are in LDS; prior reads done

        if (kt + 16 < K) {      // overlap next tile's DMA with this tile's WMMAs
            const int kn = kt + 16;
            const int nb = buf ^ 1;
            async_ld_b128(lds_off_of(&As[nb][row0][c40]), A + (size_t)(bm + row0) * K + kn + c40);
            async_ld_b128(lds_off_of(&Bs[nb][row0][c40]), B + (size_t)(bn + row0) * K + kn + c40);
            async_ld_b128(lds_off_of(&As[nb][row1][c41]), A + (size_t)(bm + row1) * K + kn + c41);
            async_ld_b128(lds_off_of(&Bs[nb][row1][c41]), B + (size_t)(bn + row1) * K + kn + c41);
        }

#pragma unroll
        for (int ks = 0; ks < 16; ks += 4) {
            v2f a;
            a.x = As[buf][wave * 16 + mr][ks + kh];
            a.y = As[buf][wave * 16 + mr][ks + kh + 1];
#pragma unroll
            for (int j = 0; j < 4; ++j) {
                v2f b;
                b.x = Bs[buf][j * 16 + mr][ks + kh];
                b.y = Bs[buf][j * 16 + mr][ks + kh + 1];
                acc[j] = wmma4(a, b, acc[j]);
            }
        }
        buf ^= 1;
    }

    // C/D layout: element r -> row = r (+8 for lanes 16-31), col = lane&15
    const int rofs = (lane >> 4) * 8;
#pragma unroll
    for (int j = 0; j < 4; ++j) {
#pragma unroll
        for (int r = 0; r < 8; ++r) {
            int gr = bm + wave * 16 + rofs + r;
            int gc = bn + j * 16 + mr;
            size_t idx = (size_t)gr * N + gc;
            float val = acc[j][r];
            if (mode == EP_STORE) {
                C[idx] = val;
            } else if (mode == EP_ADD) {
                C[idx] = val + aux[idx];
            } else if (mode == EP_SILU_MUL) {
                float g = aux[idx];
                C[idx] = (g / (1.0f + expf(-g))) * val;   // silu(g) * val
            } else { // EP_SCALE_ACC
                C[idx] += rowscale[(size_t)gr * rs_stride] * val;
            }
        }
    }
}

// ---------------------------------------------------------------------------
// RMSNorm: one block per token row of H=2048
// ---------------------------------------------------------------------------
__global__ __launch_bounds__(256)
void rmsnorm_kernel(const float* __restrict__ x, const float* __restrict__ w,
                    float* __restrict__ out)
{
    __shared__ float red[256];
    const int t = blockIdx.x;
    const float* xr = x + (size_t)t * H;
    float* orow = out + (size_t)t * H;
    float ss = 0.f;
    for (int i = threadIdx.x; i < H; i += 256) { float v = xr[i]; ss += v * v; }
    red[threadIdx.x] = ss;
    __syncthreads();
    for (int s = 128; s > 0; s >>= 1) {
        if (threadIdx.x < s) red[threadIdx.x] += red[threadIdx.x + s];
        __syncthreads();
    }
    float inv = rsqrtf(red[0] / (float)H + EPS);
    for (int i = threadIdx.x; i < H; i += 256) orow[i] = w[i] * xr[i] * inv;
}

// ---------------------------------------------------------------------------
// RoPE in-place on q [S, NH*HD] and k [S, NKV*HD]; one thread owns (d, d+32)
// ---------------------------------------------------------------------------
__global__ __launch_bounds__(256)
void rope_kernel(float* __restrict__ q, float* __restrict__ k,
                 const long long* __restrict__ pos)
{
    const int t = blockIdx.x;
    const float p = (float)pos[t];
    const float lt = logf(10000.0f);
    for (int idx = threadIdx.x; idx < NH * 32; idx += 256) {
        int h = idx >> 5, d = idx & 31;
        float ang = p * expf(-((float)d / 32.0f) * lt);
        float c = cosf(ang), s = sinf(ang);
        float* base = q + (size_t)t * (NH * HD) + h * HD;
        float x1 = base[d], x2 = base[d + 32];
        base[d]      = x1 * c - x2 * s;
        base[d + 32] = x2 * c + x1 * s;
    }
    for (int idx = threadIdx.x; idx < NKV * 32; idx += 256) {
        int h = idx >> 5, d = idx & 31;
        float ang = p * expf(-((float)d / 32.0f) * lt);
        float c = cosf(ang), s = sinf(ang);
        float* base = k + (size_t)t * KVW + h * HD;
        float x1 = base[d], x2 = base[d + 32];
        base[d]      = x1 * c - x2 * s;
        base[d + 32] = x2 * c + x1 * s;
    }
}

// ---------------------------------------------------------------------------
// Causal flash attention, GQA 4:1. Block = (head, 128 q-rows), 128 threads,
// one q row per thread; K/V streamed through LDS in 32-key tiles (hot in L2).
// ---------------------------------------------------------------------------
__global__ __launch_bounds__(128)
void attn_kernel(const float* __restrict__ q, const float* __restrict__ k,
                 const float* __restrict__ v, float* __restrict__ ctx)
{
    __shared__ float Ks[32][64];
    __shared__ float Vs[32][64];
    const int tid  = threadIdx.x;
    const int head = blockIdx.x;
    const int qblk = blockIdx.y;
    const int qr   = qblk * 128 + tid;
    const int kvh  = head >> 2;          // NH/NKV = 4
    const float scale = 0.125f;          // 1/sqrt(64)

    float qreg[64];
    const float* qrow = q + (size_t)qr * (NH * HD) + head * HD;
#pragma unroll
    for (int d4 = 0; d4 < 16; ++d4) {
        float4 t4 = *(const float4*)(qrow + d4 * 4);
        qreg[d4 * 4 + 0] = t4.x; qreg[d4 * 4 + 1] = t4.y;
        qreg[d4 * 4 + 2] = t4.z; qreg[d4 * 4 + 3] = t4.w;
    }
    float o[64];
#pragma unroll
    for (int d = 0; d < 64; ++d) o[d] = 0.f;
    float m = -1e38f, l = 0.f;

    const int ntiles = qblk * 4 + 4;     // keys up to (qblk+1)*128
    for (int kt = 0; kt < ntiles; ++kt) {
        const int key0 = kt * 32;
        for (int f = tid; f < 512; f += 128) {
            int row = f >> 4, c4 = (f & 15) << 2;
            *(float4*)&Ks[row][c4] =
                *(const float4*)(k + (size_t)(key0 + row) * KVW + kvh * HD + c4);
            *(float4*)&Vs[row][c4] =
                *(const float4*)(v + (size_t)(key0 + row) * KVW + kvh * HD + c4);
        }
        __syncthreads();

        float sreg[32];
        float smax = -1e38f;
        for (int kc = 0; kc < 32; ++kc) {
            float acc = 0.f;
#pragma unroll
            for (int d = 0; d < 64; ++d) acc += qreg[d] * Ks[kc][d];
            acc *= scale;
            if (key0 + kc > qr) acc = -1e38f;   // causal mask
            sreg[kc] = acc;
            smax = fmaxf(smax, acc);
        }
        float newm = fmaxf(m, smax);
        if (newm > -9.0e37f) {
            float c = expf(m - newm);
            float lsum = 0.f;
            for (int kc = 0; kc < 32; ++kc) {
                float pv = expf(sreg[kc] - newm);
                sreg[kc] = pv; lsum += pv;
            }
            l = l * c + lsum;
            m = newm;
            for (int d = 0; d < 64; ++d) {
                float acc = 0.f;
#pragma unroll
                for (int kc = 0; kc < 32; ++kc) acc += sreg[kc] * Vs[kc][d];
                o[d] = o[d] * c + acc;
            }
        }
        __syncthreads();
    }
    const float invl = 1.0f / l;
    float* orow = ctx + (size_t)qr * (NH * HD) + head * HD;
#pragma unroll
    for (int d = 0; d < 64; ++d) orow[d] = o[d] * invl;
}

// ---------------------------------------------------------------------------
// Router: logits = x @ wg^T (N=8), softmax, top-2 renormalized -> wfull[S,8];
// raw logits -> gate layer 0. One block per token, wave e handles expert e.
// ---------------------------------------------------------------------------
__global__ __launch_bounds__(256)
void router_kernel(const float* __restrict__ x, const float* __restrict__ wg,
                   float* __restrict__ wfull, float* __restrict__ gate0)
{
    __shared__ float red[256];
    __shared__ float lg[8];
    const int t    = blockIdx.x;
    const int e    = threadIdx.x >> 5;
    const int lane = threadIdx.x & 31;
    const float* xr = x + (size_t)t * H;
    const float* wr = wg + (size_t)e * H;
    float p = 0.f;
    for (int i = lane; i < H; i += 32) p += xr[i] * wr[i];
    red[threadIdx.x] = p;
    __syncthreads();
    for (int s = 16; s > 0; s >>= 1) {
        if (lane < s) red[threadIdx.x] += red[threadIdx.x + s];
        __syncthreads();
    }
    if (lane == 0) lg[e] = red[threadIdx.x];
    __syncthreads();
    if (threadIdx.x == 0) {
        float mx = lg[0];
        for (int i = 1; i < 8; ++i) mx = fmaxf(mx, lg[i]);
        float ex[8];
        for (int i = 0; i < 8; ++i) ex[i] = expf(lg[i] - mx);
        int i0 = 0;
        for (int i = 1; i < 8; ++i) if (ex[i] > ex[i0]) i0 = i;
        int i1 = (i0 == 0) ? 1 : 0;
        for (int i = 0; i < 8; ++i) if (i != i0 && ex[i] > ex[i1]) i1 = i;
        float wn0 = ex[i0] / (ex[i0] + ex[i1]);
        for (int i = 0; i < 8; ++i) wfull[(size_t)t * 8 + i] = 0.f;
        wfull[(size_t)t * 8 + i0] = wn0;
        wfull[(size_t)t * 8 + i1] = 1.0f - wn0;
        for (int i = 0; i < 8; ++i) gate0[(size_t)t * 8 + i] = lg[i];
    }
}

// ---------------------------------------------------------------------------
extern "C" void kernel_launch(void* const* d_in, const int* in_sizes, int n_in,
                              void* d_out, int out_size, void* d_ws, size_t ws_size,
                              hipStream_t stream)
{
    (void)in_sizes; (void)n_in; (void)out_size; (void)ws_size;

    const float*      hidden  = (const float*)d_in[0];
    const long long*  pos     = (const long long*)d_in[1];
    const float*      gate_in = (const float*)d_in[2];
    const float*      ln1     = (const float*)d_in[3];
    const float*      ln2     = (const float*)d_in[4];
    const float*      wq      = (const float*)d_in[5];
    const float*      wk      = (const float*)d_in[6];
    const float*      wv      = (const float*)d_in[7];
    const float*      wo      = (const float*)d_in[8];
    const float*      wg      = (const float*)d_in[9];
    const float*      w1      = (const float*)d_in[10];
    const float*      w2      = (const float*)d_in[11];
    const float*      w3      = (const float*)d_in[12];

    constexpr size_t SH  = (size_t)S * H;     // 4.19M
    constexpr size_t SKV = (size_t)S * KVW;   // 1.05M
    constexpr size_t SI  = (size_t)S * I;     // 8.39M

    float* ws    = (float*)d_ws;
    float* h1n    = ws;  ws += SH;
    float* qbuf   = ws;  ws += SH;
    float* kbuf   = ws;  ws += SKV;
    float* vbuf   = ws;  ws += SKV;
    float* ctx    = ws;  ws += SH;
    float* resid2 = ws;  ws += SH;
    float* h2n    = ws;  ws += SH;
    float* bufG   = ws;  ws += SI;
    float* bufH   = ws;  ws += SI;
    float* wfull  = ws;  ws += (size_t)S * E;

    // d_out = [ h : S*H f32 | position_ids : S int64 | gate_logits : NL*S*E f32 ]
    float*     out_h    = (float*)d_out;
    char*      outb     = (char*)d_out;
    long long* out_pos  = (long long*)(outb + SH * sizeof(float));
    float*     out_gate = (float*)(outb + SH * sizeof(float) + (size_t)S * sizeof(long long));

    // ---- attention block ----
    rmsnorm_kernel<<<S, 256, 0, stream>>>(hidden, ln1, h1n);
    wmma_gemm_kernel<<<dim3(H / 64,   S / 64), 128, 0, stream>>>(h1n, wq, qbuf, nullptr, nullptr, 0, S, H,   H, EP_STORE);
    wmma_gemm_kernel<<<dim3(KVW / 64, S / 64), 128, 0, stream>>>(h1n, wk, kbuf, nullptr, nullptr, 0, S, KVW, H, EP_STORE);
    wmma_gemm_kernel<<<dim3(KVW / 64, S / 64), 128, 0, stream>>>(h1n, wv, vbuf, nullptr, nullptr, 0, S, KVW, H, EP_STORE);
    rope_kernel<<<S, 256, 0, stream>>>(qbuf, kbuf, pos);
    attn_kernel<<<dim3(NH, S / 128), 128, 0, stream>>>(qbuf, kbuf, vbuf, ctx);
    // resid2 = ctx @ wo^T + hidden   (fused residual add)
    wmma_gemm_kernel<<<dim3(H / 64, S / 64), 128, 0, stream>>>(ctx, wo, resid2, hidden, nullptr, 0, S, H, H, EP_ADD);

    // ---- MoE block ----
    rmsnorm_kernel<<<S, 256, 0, stream>>>(resid2, ln2, h2n);

    hipMemcpyAsync(out_gate, gate_in, (size_t)NL * S * E * sizeof(float),
                   hipMemcpyDeviceToDevice, stream);
    router_kernel<<<S, 256, 0, stream>>>(h2n, wg, wfull, out_gate);  // overwrites layer 0

    // seed output with residual; experts accumulate on top
    hipMemcpyAsync(out_h, resid2, SH * sizeof(float), hipMemcpyDeviceToDevice, stream);
    hipMemcpyAsync(out_pos, pos, (size_t)S * sizeof(long long),
                   hipMemcpyDeviceToDevice, stream);

    for (int e = 0; e < E; ++e) {
        const float* w1e = w1 + (size_t)e * I * H;
        const float* w3e = w3 + (size_t)e * I * H;
        const float* w2e = w2 + (size_t)e * H * I;
        wmma_gemm_kernel<<<dim3(I / 64, S / 64), 128, 0, stream>>>(h2n, w1e, bufG, nullptr, nullptr, 0, S, I, H, EP_STORE);
        wmma_gemm_kernel<<<dim3(I / 64, S / 64), 128, 0, stream>>>(h2n, w3e, bufH, bufG,   nullptr, 0, S, I, H, EP_SILU_MUL);
        wmma_gemm_kernel<<<dim3(H / 64, S / 64), 128, 0, stream>>>(bufH, w2e, out_h, nullptr, wfull + e, E, S, H, I, EP_SCALE_ACC);
    }
}